// GeneralizedGraphDiffusion_18940805776195
// MI455X (gfx1250) — compile-verified
//
#include <hip/hip_runtime.h>

typedef __attribute__((ext_vector_type(2))) float v2f;
typedef __attribute__((ext_vector_type(8))) float v8f;

#define NN   5000
#define SS   8
#define DD   128
#define KCH  160   // K-range per wave slice; all slices have (kend-kstart) % 8 == 0
#define KSPL 8     // gridDim.y ; total K slices = KSPL*4 = 32 ; 32*160 = 5120 >= 5000

// ---------------------------------------------------------------- kernel 0
__global__ void ggd_zero(float* __restrict__ p, int n) {
  int i = blockIdx.x * blockDim.x + threadIdx.x;
  if (i < n) p[i] = 0.0f;
}

// ------------------------------------------------- double-buffer machinery
struct KBuf {
  v2f   t[SS];      // T_s pair for this lane's (m, k+kb)
  v2f   a;          // mask pair
  float x0[8], x1[8]; // B-fragment scalars, 8 N-tiles
};

__device__ __forceinline__ void kb_load(KBuf& b, const float* __restrict__ Trow,
                                        const float* __restrict__ arow,
                                        const float* __restrict__ x,
                                        int k, int kb, int m) {
#pragma unroll
  for (int s = 0; s < SS; ++s)
    b.t[s] = __builtin_nontemporal_load((const v2f*)(Trow + (size_t)s * NN * NN + k));
  b.a = __builtin_nontemporal_load((const v2f*)(arow + k));
  const float* xr = x + (size_t)(k + kb) * DD + m;
#pragma unroll
  for (int t = 0; t < 8; ++t) { b.x0[t] = xr[t * 16]; b.x1[t] = xr[DD + t * 16]; }
}

__device__ __forceinline__ void kb_step(const KBuf& b, const float* __restrict__ th,
                                        float rmask, v8f (&acc)[8]) {
  float s0 = th[0] * b.t[0].x;
  float s1 = th[0] * b.t[0].y;
#pragma unroll
  for (int s = 1; s < SS; ++s) {
    s0 = __builtin_fmaf(th[s], b.t[s].x, s0);
    s1 = __builtin_fmaf(th[s], b.t[s].y, s1);
  }
  v2f A;
  A.x = s0 * b.a.x * rmask;
  A.y = s1 * b.a.y * rmask;
#pragma unroll
  for (int t = 0; t < 8; ++t) {
    v2f B;
    B.x = b.x0[t];
    B.y = b.x1[t];
    acc[t] = __builtin_amdgcn_wmma_f32_16x16x4_f32(
        false, A, false, B, (short)0, acc[t], false, false);
  }
}

// ---------------------------------------------------------------- kernel 1
// mid[r, :] += ( (sum_s theta_s * T_s[r, kslice]) * a[r, kslice] ) @ x[kslice, :]
// T and a are single-use streams -> non-temporal loads (keep x resident in L2).
// Ping-pong pipelined: loads for stage i+1 issue before the WMMAs of stage i.
__global__ __launch_bounds__(128)
void ggd_diffuse_spmm(const float* __restrict__ theta,
                      const float* __restrict__ T,
                      const float* __restrict__ x,
                      const float* __restrict__ a,
                      float* __restrict__ mid) {
  __shared__ float red[4][16 * DD];

  const int lane = threadIdx.x & 31;
  const int wave = threadIdx.x >> 5;
  const int r0   = blockIdx.x * 16;
  const int m    = lane & 15;
  const int kb   = (lane >> 4) << 1;          // 0 for lanes 0-15, 2 for lanes 16-31
  const int row  = r0 + m;
  const int rowc = row < NN ? row : NN - 1;   // clamp (keep EXEC full for WMMA)
  const float rmask = row < NN ? 1.0f : 0.0f;

  float th[SS];
#pragma unroll
  for (int s = 0; s < SS; ++s) th[s] = theta[s];

  const int kslice = blockIdx.y * 4 + wave;
  const int kstart = kslice * KCH;
  const int kend   = (kstart + KCH) < NN ? (kstart + KCH) : NN;
  // every slice: kend - kstart in {160, 40}, both divisible by 8 (pairs of k-steps)

  v8f acc[8];
#pragma unroll
  for (int t = 0; t < 8; ++t) acc[t] = {};

  const float* Trow = T + (size_t)rowc * NN + kb;   // + s*NN*NN + k per access
  const float* arow = a + (size_t)rowc * NN + kb;

  KBuf b0, b1;
  kb_load(b0, Trow, arow, x, kstart, kb, m);        // prologue

  int k = kstart;
  for (; k + 8 < kend; k += 8) {                    // main: unconditional prefetch
    kb_load(b1, Trow, arow, x, k + 4, kb, m);       //   loads in flight ...
    kb_step(b0, th, rmask, acc);                    //   ... during these WMMAs
    kb_load(b0, Trow, arow, x, k + 8, kb, m);
    kb_step(b1, th, rmask, acc);
  }
  // epilogue: exactly 2 k-steps remain
  kb_load(b1, Trow, arow, x, k + 4, kb, m);
  kb_step(b0, th, rmask, acc);
  kb_step(b1, th, rmask, acc);

  // dump partial 16x128 tile to LDS  (C layout: VGPR j -> M = j + 8*(lane>=16))
  const int mlo = (lane >> 4) << 3;
#pragma unroll
  for (int t = 0; t < 8; ++t)
#pragma unroll
    for (int j = 0; j < 8; ++j)
      red[wave][(mlo + j) * DD + t * 16 + m] = acc[t][j];

  __syncthreads();

  // cross-wave reduction + atomic add into mid
  const int tid = threadIdx.x;
#pragma unroll 4
  for (int i = 0; i < 16; ++i) {
    const int e  = tid * 16 + i;              // 128 threads * 16 = 2048 elements
    const float v = red[0][e] + red[1][e] + red[2][e] + red[3][e];
    const int rr = e >> 7;
    if (r0 + rr < NN) {
      __hip_atomic_fetch_add(&mid[(size_t)(r0 + rr) * DD + (e & (DD - 1))], v,
                             __ATOMIC_RELAXED, __HIP_MEMORY_SCOPE_AGENT);
    }
  }
}

// ---------------------------------------------------------------- kernel 2
// out = PReLU(mid) @ W^T + b      (PReLU fused into A-fragment construction)
__global__ __launch_bounds__(128)
void ggd_prelu_linear(const float* __restrict__ mid,
                      const float* __restrict__ alpha,
                      const float* __restrict__ W,
                      const float* __restrict__ b,
                      float* __restrict__ out) {
  const int lane = threadIdx.x & 31;
  const int wave = threadIdx.x >> 5;
  const int r0   = blockIdx.x * 16;
  const int m    = lane & 15;
  const int kb   = (lane >> 4) << 1;
  const int row  = r0 + m;
  const int rowc = row < NN ? row : NN - 1;

  v8f acc[2];
  acc[0] = {};
  acc[1] = {};
  const int t0 = wave * 2;   // each wave owns 2 of the 8 N-tiles

  for (int k = 0; k < DD; k += 4) {
    const v2f mv = *(const v2f*)(mid + (size_t)rowc * DD + k + kb);
    const v2f al = *(const v2f*)(alpha + k + kb);
    v2f A;
    A.x = mv.x >= 0.0f ? mv.x : al.x * mv.x;
    A.y = mv.y >= 0.0f ? mv.y : al.y * mv.y;
#pragma unroll
    for (int t = 0; t < 2; ++t) {
      const int col = (t0 + t) * 16 + m;
      const v2f B = *(const v2f*)(W + (size_t)col * DD + k + kb); // W^T col == W row
      acc[t] = __builtin_amdgcn_wmma_f32_16x16x4_f32(
          false, A, false, B, (short)0, acc[t], false, false);
    }
  }

  const int mlo = (lane >> 4) << 3;
#pragma unroll
  for (int t = 0; t < 2; ++t) {
    const int col = (t0 + t) * 16 + m;
    const float bias = b[col];
#pragma unroll
    for (int j = 0; j < 8; ++j) {
      const int rr = r0 + mlo + j;
      if (rr < NN) out[(size_t)rr * DD + col] = acc[t][j] + bias;
    }
  }
}

// ---------------------------------------------------------------- launch
extern "C" void kernel_launch(void* const* d_in, const int* in_sizes, int n_in,
                              void* d_out, int out_size, void* d_ws, size_t ws_size,
                              hipStream_t stream) {
  const float* theta = (const float*)d_in[0];  // [8]
  const float* T     = (const float*)d_in[1];  // [8,5000,5000]
  const float* x     = (const float*)d_in[2];  // [5000,128]
  const float* a     = (const float*)d_in[3];  // [5000,5000]
  const float* alpha = (const float*)d_in[4];  // [128]
  const float* W     = (const float*)d_in[5];  // [128,128]
  const float* b     = (const float*)d_in[6];  // [128]
  float* out = (float*)d_out;                  // [5000,128]
  float* mid = (float*)d_ws;                   // [5000,128] scratch accumulator

  const int midN = NN * DD;
  ggd_zero<<<(midN + 255) / 256, 256, 0, stream>>>(mid, midN);

  dim3 grid1((NN + 15) / 16, KSPL);            // 313 row tiles x 8 K splits
  ggd_diffuse_spmm<<<grid1, 128, 0, stream>>>(theta, T, x, a, mid);

  ggd_prelu_linear<<<(NN + 15) / 16, 128, 0, stream>>>(mid, alpha, W, b, out);
}